// FilteredCrossEntropyLoss_24885040513628
// MI455X (gfx1250) — compile-verified
//
#include <hip/hip_runtime.h>
#include <stdint.h>

// ---------------------------------------------------------------------------
// FilteredCrossEntropyLoss — MI455X (gfx1250)
// loss_row = H(t) + LSE(pred_row) - dot(filter, pred_row window)
// Bandwidth-bound (256MB single pass): stream pred via TDM tensor_load_to_lds,
// double-buffered in LDS; 16-lane half-wave per row (wave32); deterministic
// two-pass reduction.
// ---------------------------------------------------------------------------

#define NB        1024          // blocks
#define TPB_TILES 16            // tiles per block
#define ROWS      64            // rows per tile
#define CLS       64            // classes
#define TILE_ELEMS (ROWS * CLS) // 4096 floats = 16 KB
#define PASSES    (ROWS / 16)   // 16 half-wave groups per 256-thread block

typedef unsigned int u32x4 __attribute__((ext_vector_type(4)));
typedef int          i32x4 __attribute__((ext_vector_type(4)));
typedef int          i32x8 __attribute__((ext_vector_type(8)));

// Σ w log w constants for the three filter shapes
#define H_EDGE3 (-0.93763697f)  // .6,.25,.15
#define H_EDGE4 (-1.28992198f)  // .25,.4,.25,.1
#define H_ELSE  (-1.50336325f)  // .1,.23,.34,.23,.1

// Issue one TDM transfer: TILE_ELEMS contiguous f32 from global -> LDS.
// D# layout per CDNA5 ISA §8 (group0 128b, group1 256b; groups 2/3 unused).
__device__ __forceinline__ void tdm_load_tile(const float* gsrc, uint32_t lds_off) {
  const uint64_t ga = (uint64_t)(uintptr_t)gsrc;
  u32x4 d0 = {0u, 0u, 0u, 0u};
  d0[0] = 1u;                                   // count=1 (valid user descriptor)
  d0[1] = lds_off;                              // lds_addr (bytes, WG-relative)
  d0[2] = (uint32_t)ga;                         // global_addr[31:0]
  d0[3] = (uint32_t)((ga >> 32) & 0x01FFFFFFu)  // global_addr[56:32]
        | (2u << 30);                           // type = 2 ("image")
  i32x8 d1 = {0, 0, 0, 0, 0, 0, 0, 0};
  d1[0] = (2 << 16);                            // data_size = 4 bytes; mask=0
  d1[1] = (int)((TILE_ELEMS & 0xFFFF) << 16);   // tensor_dim0[15:0] @ bits 63:48
  d1[2] = (TILE_ELEMS >> 16) | (1 << 16);       // tensor_dim0[31:16] | tensor_dim1=1
  d1[3] = (int)((unsigned)TILE_ELEMS << 16);    // tile_dim0 = 4096 @ bits 127:112
  d1[4] = 1;                                    // tile_dim1 = 1, tile_dim2 = 0
  d1[5] = TILE_ELEMS;                           // tensor_dim0_stride[31:0]
  i32x4 z = {0, 0, 0, 0};
#if __clang_major__ >= 23
  i32x8 z8 = {0, 0, 0, 0, 0, 0, 0, 0};
  __builtin_amdgcn_tensor_load_to_lds(d0, d1, z, z, z8, 0);
#else
  __builtin_amdgcn_tensor_load_to_lds(d0, d1, z, z, 0);
#endif
}

__global__ __launch_bounds__(256) void fce_main_kernel(
    const float* __restrict__ pred,
    const int*   __restrict__ target,
    float*       __restrict__ partial) {
  __shared__ __align__(16) float tile[2][TILE_ELEMS];   // 32 KB double buffer

  const int tid   = threadIdx.x;
  const int wave  = tid >> 5;
  const int grp   = tid >> 4;     // 16 half-wave groups, one row each per pass
  const int lane  = tid & 15;     // lane within group: classes 4*lane..4*lane+3
  const int tbase = blockIdx.x * TPB_TILES;

  // Prologue: wave0 kicks off tile 0 via TDM.
  if (wave == 0) {
    tdm_load_tile(pred + (size_t)tbase * TILE_ELEMS,
                  (uint32_t)(uintptr_t)&tile[0][0]);
  }

  float acc = 0.0f;
  for (int i = 0; i < TPB_TILES; ++i) {
    const int buf = i & 1;
    if (wave == 0) __builtin_amdgcn_s_wait_tensorcnt(0);  // tile i landed in LDS
    __syncthreads();  // publish tile i; also guarantees compute(i-1) finished,
                      // so buf^1 is free to overwrite below.
    if (wave == 0 && (i + 1) < TPB_TILES) {
      tdm_load_tile(pred + (size_t)(tbase + i + 1) * TILE_ELEMS,
                    (uint32_t)(uintptr_t)&tile[buf ^ 1][0]);  // overlap with compute
    }

    const size_t rowbase = (size_t)(tbase + i) * ROWS;
#pragma unroll
    for (int p = 0; p < PASSES; ++p) {
      const int r = p * 16 + grp;
      const float4 v = *(const float4*)&tile[buf][r * CLS + lane * 4];
      const int t = target[rowbase + r];   // uniform within the 16-lane group

      // Contiguous filter window: start p0, weights w0..w4 (unused = 0), entropy Hc.
      int p0; float w0, w1, w2, w3, w4, Hc;
      if (t == 0)            { p0 = 0;  w0=0.60f; w1=0.25f; w2=0.15f; w3=0.f;   w4=0.f;   Hc=H_EDGE3; }
      else if (t == 1)       { p0 = 0;  w0=0.25f; w1=0.40f; w2=0.25f; w3=0.10f; w4=0.f;   Hc=H_EDGE4; }
      else if (t == CLS - 2) { p0 = 60; w0=0.10f; w1=0.25f; w2=0.40f; w3=0.25f; w4=0.f;   Hc=H_EDGE4; }
      else if (t == CLS - 1) { p0 = 61; w0=0.15f; w1=0.25f; w2=0.60f; w3=0.f;   w4=0.f;   Hc=H_EDGE3; }
      else { // torch roll: element j lands at (padding + j + t - 2) mod C = (t+27+j) & 63
        p0 = (t + 27) & 63;
        w0=0.10f; w1=0.23f; w2=0.34f; w3=0.23f; w4=0.10f; Hc=H_ELSE;
      }

      // Row max across 16 lanes (LSE stabilization).
      float m = fmaxf(fmaxf(v.x, v.y), fmaxf(v.z, v.w));
#pragma unroll
      for (int s = 1; s < 16; s <<= 1) m = fmaxf(m, __shfl_xor(m, s, 16));

      // exp-sum partial + windowed-dot partial for this lane's 4 classes.
      float e = __expf(v.x - m) + __expf(v.y - m) + __expf(v.z - m) + __expf(v.w - m);
      const float vv[4] = {v.x, v.y, v.z, v.w};
      float dot = 0.0f;
      const int c0 = lane * 4;
#pragma unroll
      for (int k = 0; k < 4; ++k) {
        const int idx = ((c0 + k) - p0) & 63;  // distance into the (wrapping) window
        const float w = (idx == 0) ? w0 : (idx == 1) ? w1 : (idx == 2) ? w2
                      : (idx == 3) ? w3 : (idx == 4) ? w4 : 0.0f;
        dot = fmaf(w, vv[k], dot);
      }
#pragma unroll
      for (int s = 1; s < 16; s <<= 1) {
        e   += __shfl_xor(e,   s, 16);
        dot += __shfl_xor(dot, s, 16);
      }
      if (lane == 0) acc += Hc + m + __logf(e) - dot;   // H + LSE - <tp,pred>
    }
  }

  // Deterministic block reduction (reuse tile LDS after final barrier).
  __syncthreads();
  float a = acc;
#pragma unroll
  for (int s = 1; s < 32; s <<= 1) a += __shfl_xor(a, s, 32);
  if ((tid & 31) == 0) tile[0][wave] = a;
  __syncthreads();
  if (tid == 0) {
    float s = 0.0f;
#pragma unroll
    for (int w = 0; w < 8; ++w) s += tile[0][w];
    partial[blockIdx.x] = s;
  }
}

__global__ __launch_bounds__(256) void fce_reduce_kernel(
    const float* __restrict__ partial, float* __restrict__ out) {
  __shared__ float red[256];
  const int tid = threadIdx.x;
  float s = 0.0f;
  for (int i = tid; i < NB; i += 256) s += partial[i];  // fixed order
  red[tid] = s;
  __syncthreads();
  for (int off = 128; off > 0; off >>= 1) {
    if (tid < off) red[tid] += red[tid + off];
    __syncthreads();
  }
  if (tid == 0) out[0] = red[0] * (1.0f / 1048576.0f);  // batchmean
}

extern "C" void kernel_launch(void* const* d_in, const int* in_sizes, int n_in,
                              void* d_out, int out_size, void* d_ws, size_t ws_size,
                              hipStream_t stream) {
  (void)in_sizes; (void)n_in; (void)out_size; (void)ws_size;
  const float* pred   = (const float*)d_in[0];
  const int*   target = (const int*)d_in[1];   // harness: integer -> const int*
  float* partial = (float*)d_ws;               // NB floats of scratch
  float* out     = (float*)d_out;
  fce_main_kernel<<<NB, 256, 0, stream>>>(pred, target, partial);
  fce_reduce_kernel<<<1, 256, 0, stream>>>(partial, out);
}